// vLoss_36112085024869
// MI455X (gfx1250) — compile-verified
//
#include <hip/hip_runtime.h>
#include <hip/hip_bf16.h>
#include <math.h>

// Problem constants (reference: N,K,C,W,H = 4,4,256,128,128)
#define NN 4
#define KK 4
#define CC 256
#define PP 16384   // W*H

typedef __attribute__((ext_vector_type(2))) float v2f;
typedef __attribute__((ext_vector_type(8))) float v8f;

// ---------------------------------------------------------------------------
// Kernel 1: means[n,k,c] = (1/P) * sum_p output[n,c,p] * gt[n,k,p]
// Stored K-REVERSED (reference does means[:, ::-1, :]) and zero-padded to 16
// rows so kernel 2's WMMA A-matrix needs no lane predication (EXEC all-1s).
// One block per (c, n); 256 threads stream the 16384-pixel row once.
// ---------------------------------------------------------------------------
__global__ void vloss_means_kernel(const float* __restrict__ out,
                                   const int* __restrict__ target,
                                   float* __restrict__ meansR) {
    const int c   = blockIdx.x;
    const int n   = blockIdx.y;
    const int tid = threadIdx.x;

    const float* src = out + ((size_t)n * CC + c) * PP;
    const int*   tg  = target + (size_t)n * KK * PP;

    float s0 = 0.f, s1 = 0.f, s2 = 0.f, s3 = 0.f;
    for (int p = tid; p < PP; p += 256) {
        float v = src[p];
        s0 += v * (float)tg[p];
        s1 += v * (float)tg[PP + p];
        s2 += v * (float)tg[2 * PP + p];
        s3 += v * (float)tg[3 * PP + p];
    }
    // wave32 reduction
    for (int off = 16; off >= 1; off >>= 1) {
        s0 += __shfl_xor(s0, off, 32);
        s1 += __shfl_xor(s1, off, 32);
        s2 += __shfl_xor(s2, off, 32);
        s3 += __shfl_xor(s3, off, 32);
    }
    __shared__ float wsum[8][4];
    const int wave = tid >> 5, lane = tid & 31;
    if (lane == 0) {
        wsum[wave][0] = s0; wsum[wave][1] = s1;
        wsum[wave][2] = s2; wsum[wave][3] = s3;
    }
    __syncthreads();
    if (tid < 4) {
        float t = 0.f;
        for (int w = 0; w < 8; ++w) t += wsum[w][tid];
        // reversed k, scaled by 1/P
        meansR[((size_t)n * 16 + (3 - tid)) * CC + c] = t * (1.0f / (float)PP);
    } else if (tid < 16) {
        meansR[((size_t)n * 16 + tid) * CC + c] = 0.0f;  // pad rows 4..15
    }
}

// ---------------------------------------------------------------------------
// Kernel 2: fused x2 / xm (via V_WMMA_F32_16X16X4_F32) / dist / argmin-pred.
// 8 waves per block, one 16-pixel tile per wave. Per C-chunk of 4:
//   A (16x4 f32, meansR rows = padded k): lanes 0-15 hold M=row, VGPR0 holds
//     Kdim {0 | 2} by lane-half, VGPR1 holds Kdim {1 | 3}.
//   B (4x16 f32, output): mirrored layout — lane = pixel column, lane-half
//     selects channel pair (c+2h, c+2h+1).
// acc (v8f D): lanes 0-15, VGPR r = xm[k=r][p0+lane] for r<4 (rows 4..15 = 0).
// x2 accumulated from the same B registers; halves combined with a lane XOR-16.
// pred (argmin label, largest index on ties — reference semantics) is STORED
// to the workspace: this is the memory anchor that keeps the whole WMMA chain
// live (the downstream loss folds to a constant at compile time otherwise).
// ---------------------------------------------------------------------------
__global__ void vloss_dist_kernel(const float* __restrict__ out,
                                  const float* __restrict__ meansR,
                                  int* __restrict__ predBuf) {
    __shared__ float m2part[256];
    __shared__ float m2s[4];

    const int tid   = threadIdx.x;
    const int tile0 = blockIdx.x * 8;     // 8 waves/block; 1024 tiles per n (8 | 1024)
    const int nb    = tile0 >> 10;        // whole block shares one n

    // m2[k] = sum_c meansR[k][c]^2 (deterministic tree, no atomics)
    {
        const int k = tid >> 6, j = tid & 63;
        const float* mr = meansR + ((size_t)nb * 16 + k) * CC + j * 4;
        float p = mr[0] * mr[0] + mr[1] * mr[1] + mr[2] * mr[2] + mr[3] * mr[3];
        m2part[tid] = p;
    }
    __syncthreads();
    if (tid < 4) {
        float s = 0.f;
        for (int i = 0; i < 64; ++i) s += m2part[tid * 64 + i];
        m2s[tid] = s;
    }
    __syncthreads();

    const int wave = tid >> 5, lane = tid & 31;
    const int tile = tile0 + wave;
    const int n    = tile >> 10;
    const int p0   = (tile & 1023) << 4;

    const int row  = lane & 15;   // A: M row / B: pixel column
    const int half = lane >> 4;   // selects Kdim pair {0,1} vs {2,3}

    const float* mrow  = meansR + ((size_t)n * 16 + row) * CC;
    const float* bbase = out + ((size_t)n * CC + 2 * half) * PP + p0 + row;

    v8f acc = {};
    float x2p = 0.f;
    for (int c = 0; c < CC; c += 4) {
        v2f A, B;
        A.x = mrow[c + 2 * half];
        A.y = mrow[c + 2 * half + 1];
        const float* bp = bbase + (size_t)c * PP;
        __builtin_prefetch(bp + (size_t)16 * PP, 0, 1);   // 4 chunks ahead -> global_prefetch_b8
        float b0 = bp[0];
        float b1 = bp[PP];
        x2p += b0 * b0 + b1 * b1;
        B.x = b0; B.y = b1;
        // D = A(16x4) x B(4x16) + D   :: v_wmma_f32_16x16x4_f32
        acc = __builtin_amdgcn_wmma_f32_16x16x4_f32(
            false, A, false, B, (short)0, acc, false, false);
    }
    // combine channel halves: lane j + lane j+16 cover all 256 channels
    const float x2full = x2p + __shfl_xor(x2p, 16, 32);

    if (lane < 16) {
        float best = __FLT_MAX__;
        int   pred = 0;
        #pragma unroll
        for (int k = 0; k < 4; ++k) {
            float xm   = acc[k];                       // D row k, col p0+lane
            float sq   = x2full - 2.0f * xm + m2s[k];
            float dist = sqrtf(fmaxf(sq, 0.0f) * (1.0f / (float)CC));
            if (dist <= best) { best = dist; pred = k; }  // largest k on ties
        }
        predBuf[(size_t)n * PP + p0 + lane] = pred;    // memory anchor
    }
}

// ---------------------------------------------------------------------------
// Kernel 3: loss = -sum_valid(log_softmax(pred over singleton axis)) / P / N
// Single block; deterministic tree reduction.
// ---------------------------------------------------------------------------
__global__ void vloss_final_kernel(const int* __restrict__ predBuf,
                                   const int* __restrict__ tvc,
                                   float* __restrict__ outv) {
    __shared__ float red[256];
    const int tid = threadIdx.x;
    float s = 0.f;
    for (int i = tid; i < NN * PP; i += 256) {
        float logit = (float)predBuf[i];
        float m     = logit;
        float lse   = m + logf(expf(logit - m));      // logsumexp over size-1 axis
        float logp  = logit - lse;
        s += (tvc[i] != 255) ? logp : 0.0f;
    }
    red[tid] = s;
    __syncthreads();
    for (int off = 128; off >= 1; off >>= 1) {
        if (tid < off) red[tid] += red[tid + off];
        __syncthreads();
    }
    if (tid == 0) outv[0] = -red[0] / (float)PP / (float)NN;
}

extern "C" void kernel_launch(void* const* d_in, const int* in_sizes, int n_in,
                              void* d_out, int out_size, void* d_ws, size_t ws_size,
                              hipStream_t stream) {
    const float* output = (const float*)d_in[0];   // (N,C,W,H) f32
    const int*   target = (const int*)d_in[1];     // (N,K,W,H) i32
    const int*   tvc    = (const int*)d_in[2];     // (N,1,W,H) i32
    float*       outv   = (float*)d_out;

    float* meansR  = (float*)d_ws;                       // N*16*C floats (64 KB)
    int*   predBuf = (int*)(meansR + (size_t)NN * 16 * CC);  // N*P ints (256 KB)

    // Pass 1: means (reads output once, 67 MB)
    vloss_means_kernel<<<dim3(CC, NN), 256, 0, stream>>>(output, target, meansR);
    // Pass 2: fused x2/xm(WMMA)/dist/pred (reads output once more, 67 MB)
    const int tiles = NN * (PP / 16);              // 4096 tiles, 8 waves/block
    vloss_dist_kernel<<<tiles / 8, 256, 0, stream>>>(output, meansR, predBuf);
    // Finalize loss from pred + valid mask
    vloss_final_kernel<<<1, 256, 0, stream>>>(predBuf, tvc, outv);

    (void)in_sizes; (void)n_in; (void)out_size; (void)ws_size;
}